// EGATDP_42863773614894
// MI455X (gfx1250) — compile-verified
//
#include <hip/hip_runtime.h>
#include <hip/hip_bf16.h>
#include <math.h>

#define NN 4096
#define EE 131072
#define PP 3
#define FINN 128
#define KSPLIT 8
#define KCH (NN / KSPLIT)  // 512

typedef __attribute__((ext_vector_type(2))) float v2f;
typedef __attribute__((ext_vector_type(8))) float v8f;

// ---------------- helpers ----------------

__device__ __forceinline__ float eluf(float v) { return v > 0.f ? v : expm1f(v); }

__device__ __forceinline__ float atomicMaxF(float* addr, float val) {
  int* ai = (int*)addr;
  int old = __float_as_int(*addr);
  while (__int_as_float(old) < val) {
    int prev = atomicCAS(ai, old, __float_as_int(val));
    if (prev == old) break;
    old = prev;
  }
  return __int_as_float(old);
}

__device__ __forceinline__ float waveReduceSum(float v) {
#pragma unroll
  for (int off = 16; off > 0; off >>= 1) v += __shfl_down(v, off, 32);
  return v;
}

__device__ __forceinline__ float blockReduceSum256(float v) {
  __shared__ float sm[8];
  int lane = threadIdx.x & 31, w = threadIdx.x >> 5;
  v = waveReduceSum(v);
  if (lane == 0) sm[w] = v;
  __syncthreads();
  if (w == 0) {
    v = (lane < (int)(blockDim.x >> 5)) ? sm[lane] : 0.f;
    v = waveReduceSum(v);
  }
  return v;  // valid in thread 0
}

// ---------------- fills ----------------

__global__ void k_fill(float* __restrict__ p, float v, int n) {
  int i = blockIdx.x * blockDim.x + threadIdx.x;
  if (i < n) p[i] = v;
}

__global__ void k_filli(int* __restrict__ p, int v, int n) {
  int i = blockIdx.x * blockDim.x + threadIdx.x;
  if (i < n) p[i] = v;
}

// ---------------- CSR-by-destination build (shared by both EGAT layers) ----------------

__global__ void k_count(const int* __restrict__ dst, int* __restrict__ deg) {
  int e = blockIdx.x * blockDim.x + threadIdx.x;
  if (e < EE) atomicAdd(&deg[dst[e]], 1);
}

__global__ void k_scan4096(const int* __restrict__ deg, int* __restrict__ offs,
                           int* __restrict__ cursor) {
  // single block, 256 threads, exclusive scan of 4096 ints
  __shared__ int part[256];
  __shared__ int pref[257];
  int t = threadIdx.x;
  int base = t * 16;
  int local[16];
  int s = 0;
#pragma unroll
  for (int i = 0; i < 16; ++i) { local[i] = s; s += deg[base + i]; }
  part[t] = s;
  __syncthreads();
  if (t == 0) {
    int acc = 0;
    for (int i = 0; i < 256; ++i) { pref[i] = acc; acc += part[i]; }
    pref[256] = acc;
  }
  __syncthreads();
  int off = pref[t];
#pragma unroll
  for (int i = 0; i < 16; ++i) {
    offs[base + i] = off + local[i];
    cursor[base + i] = off + local[i];
  }
  if (t == 0) offs[NN] = pref[256];
}

__global__ void k_scatter(const int* __restrict__ dst, int* __restrict__ cursor,
                          int* __restrict__ eperm) {
  int e = blockIdx.x * blockDim.x + threadIdx.x;
  if (e >= EE) return;
  int pos = atomicAdd(&cursor[dst[e]], 1);
  eperm[pos] = e;
}

// ---------------- GEMM1: h1 = x @ W1 (4096x128 @ 128x16), WMMA f32 16x16x4 ----------------

__global__ void __launch_bounds__(32) k_gemm1_wmma(const float* __restrict__ x,
                                                   const float* __restrict__ W1,
                                                   float* __restrict__ h1) {
  const int lane = threadIdx.x;
  const int rowb = blockIdx.x * 16;
  const int m = lane & 15;
  const int koff = (lane >> 4) << 1;  // lanes 0-15: K=0,1 ; lanes 16-31: K=2,3
  v8f c = {};
  for (int k = 0; k < FINN; k += 4) {
    v2f a = *(const v2f*)(x + (size_t)(rowb + m) * FINN + k + koff);
    v2f b;
    b.x = W1[(k + koff + 0) * 16 + m];
    b.y = W1[(k + koff + 1) * 16 + m];
    c = __builtin_amdgcn_wmma_f32_16x16x4_f32(false, a, false, b, (short)0, c, false, false);
  }
  const int rbase = (lane >> 4) << 3;  // C: vgpr r -> row r (lanes 0-15) / r+8 (lanes 16-31)
#pragma unroll
  for (int r = 0; r < 8; ++r) h1[(size_t)(rowb + rbase + r) * 16 + m] = c[r];
}

// ---------------- per-node attention projections ----------------

__global__ void k_nodeproj(const float* __restrict__ h, int F,
                           const float* __restrict__ a_s, const float* __restrict__ a_d,
                           float* __restrict__ ssrc, float* __restrict__ sdst) {
  int n = blockIdx.x * blockDim.x + threadIdx.x;
  if (n >= NN) return;
  float s = 0.f, d = 0.f;
  for (int f = 0; f < F; ++f) {
    float hv = h[(size_t)n * F + f];
    s = fmaf(hv, a_s[f], s);
    d = fmaf(hv, a_d[f], d);
  }
  ssrc[n] = s;
  sdst[n] = d;
}

// ---------------- edge score + segment max ----------------

__global__ void k_score_max(const int* __restrict__ src, const int* __restrict__ dst,
                            const float* __restrict__ ssrc, const float* __restrict__ sdst,
                            float* __restrict__ score, float* __restrict__ smax) {
  int e = blockIdx.x * blockDim.x + threadIdx.x;
  if (e >= EE) return;
  float s = ssrc[src[e]] + sdst[dst[e]];
  s = s > 0.f ? s : 0.2f * s;  // leaky_relu(0.2)
  score[e] = s;
  atomicMaxF(&smax[dst[e]], s);
}

// ---------------- exp + segment sum (denominator) ----------------

__global__ void k_expden(const int* __restrict__ dst, float* __restrict__ ex,
                         const float* __restrict__ smax, float* __restrict__ den) {
  int e = blockIdx.x * blockDim.x + threadIdx.x;
  if (e >= EE) return;
  float v = expf(ex[e] - smax[dst[e]]);
  ex[e] = v;
  atomicAdd(&den[dst[e]], v);
}

// ---------------- new edge features: ae = alpha * prev_edge_feat ----------------

__global__ void k_ae(const int* __restrict__ dst, const float* __restrict__ ex,
                     const float* __restrict__ den, const float* __restrict__ prev,
                     float* __restrict__ ae) {
  int tid = blockIdx.x * blockDim.x + threadIdx.x;
  if (tid >= EE * PP) return;
  int e = tid / PP;
  float alpha = ex[e] / den[dst[e]];
  ae[tid] = alpha * prev[tid];
}

// ---------------- gather messages: out[n][p*F+f] = Sum_{e in in(n)} ae[e][p]*h[src_e][f] ----

__global__ void k_gather_msg(const int* __restrict__ offs, const int* __restrict__ eperm,
                             const int* __restrict__ src, const float* __restrict__ ae,
                             const float* __restrict__ h, float* __restrict__ out, int F) {
  int n = blockIdx.x;
  int PF = PP * F;
  int pf = threadIdx.x;
  if (pf >= PF) return;
  int p = pf / F, f = pf - p * F;
  int b = offs[n], e2 = offs[n + 1];
  float acc = 0.f;
  for (int i = b; i < e2; ++i) {
    int e = eperm[i];
    acc = fmaf(ae[(size_t)e * PP + p], h[(size_t)src[e] * F + f], acc);
  }
  out[(size_t)n * PF + pf] = acc;
}

// ---------------- elementwise ELU in place ----------------

__global__ void k_elu(float* __restrict__ p, int n) {
  int i = blockIdx.x * blockDim.x + threadIdx.x;
  if (i < n) p[i] = eluf(p[i]);
}

// ---------------- GEMM2: h2 = out1 @ W2 (4096x48 @ 48x4), tiny ----------------

__global__ void k_gemm2(const float* __restrict__ a, const float* __restrict__ W2,
                        float* __restrict__ h2) {
  int tid = blockIdx.x * blockDim.x + threadIdx.x;
  if (tid >= NN * 4) return;
  int n = tid >> 2, j = tid & 3;
  float acc = 0.f;
  for (int k = 0; k < 48; ++k) acc = fmaf(a[(size_t)n * 48 + k], W2[k * 4 + j], acc);
  h2[tid] = acc;
}

// ---------------- row softmax of S1 -> s1, fused entropy accumulation ----------------

__global__ void k_softmax_s1(const float* __restrict__ S1, float* __restrict__ s1,
                             float* __restrict__ entAccum) {
  int n = blockIdx.x * blockDim.x + threadIdx.x;
  if (n >= NN) return;
  float v[16];
  float mx = -3.4e38f;
#pragma unroll
  for (int c = 0; c < 16; ++c) { v[c] = S1[n * 16 + c]; mx = fmaxf(mx, v[c]); }
  float sum = 0.f;
#pragma unroll
  for (int c = 0; c < 16; ++c) { v[c] = expf(v[c] - mx); sum += v[c]; }
  float inv = 1.f / sum, ent = 0.f;
#pragma unroll
  for (int c = 0; c < 16; ++c) {
    float sv = v[c] * inv;
    s1[n * 16 + c] = sv;
    ent -= sv * logf(sv + 1e-15f);
  }
  atomicAdd(entAccum, ent);
}

__global__ void k_transpose_s1(const float* __restrict__ s1, float* __restrict__ s1T) {
  int tid = blockIdx.x * blockDim.x + threadIdx.x;
  if (tid >= NN * 16) return;
  int n = tid >> 4, c = tid & 15;
  s1T[(size_t)c * NN + n] = s1[tid];
}

// ---------------- THE big kernel: single pass over adj (201 MB) ----------------
// Tpart[kc][p] += adj_p[:, kc-chunk] @ s1[kc-chunk, :] via V_WMMA_F32_16X16X4_F32,
// fused Sum(adj^2). K split 8-way -> 6144 independent waves for HBM latency hiding.

__global__ void __launch_bounds__(32) k_adj_pool(const float* __restrict__ adj,
                                                 const float* __restrict__ s1T,   // [16][NN]
                                                 float* __restrict__ Tpart,       // [KSPLIT][PP][NN][16]
                                                 float* __restrict__ adjsq) {
  const int lane = threadIdx.x;
  const int rowb = blockIdx.x * 16;
  const int p = blockIdx.y;
  const int kc = blockIdx.z;
  const int m = lane & 15;
  const int koff = (lane >> 4) << 1;
  const float* arow = adj + ((size_t)p * NN + rowb + m) * NN + (size_t)kc * KCH + koff;
  const float* brow = s1T + (size_t)m * NN + (size_t)kc * KCH + koff;
  v8f c = {};
  float sq = 0.f;
#pragma unroll 8
  for (int k = 0; k < KCH; k += 4) {
    v2f a = *(const v2f*)(arow + k);
    v2f b = *(const v2f*)(brow + k);
    c = __builtin_amdgcn_wmma_f32_16x16x4_f32(false, a, false, b, (short)0, c, false, false);
    sq = fmaf(a.x, a.x, sq);
    sq = fmaf(a.y, a.y, sq);
  }
  const int rbase = (lane >> 4) << 3;
  float* tb = Tpart + (((size_t)kc * PP + p) * NN + rowb) * 16;
#pragma unroll
  for (int r = 0; r < 8; ++r) tb[(size_t)(rbase + r) * 16 + m] = c[r];
  sq = waveReduceSum(sq);
  if (lane == 0) atomicAdd(adjsq, sq);
}

// ---------------- deterministic K-chunk reduction: T = Sum_kc Tpart[kc] ----------------

__global__ void k_reduceT(const float* __restrict__ Tpart, float* __restrict__ T) {
  int idx = blockIdx.x * blockDim.x + threadIdx.x;
  if (idx >= PP * NN * 16) return;
  float a = 0.f;
#pragma unroll
  for (int kc = 0; kc < KSPLIT; ++kc) a += Tpart[(size_t)kc * (PP * NN * 16) + idx];
  T[idx] = a;
}

// ---------------- cross term: Sum_{p,i,c} s1[i][c] * T_p[i][c] ----------------

__global__ void k_cross(const float* __restrict__ s1, const float* __restrict__ T,
                        float* __restrict__ cross) {
  int idx = blockIdx.x * blockDim.x + threadIdx.x;  // over PP*NN*16
  float v = 0.f;
  if (idx < PP * NN * 16) {
    int rem = idx & (NN * 16 - 1);  // NN*16 = 65536, power of two
    v = s1[rem] * T[idx];
  }
  v = blockReduceSum256(v);
  if (threadIdx.x == 0) atomicAdd(cross, v);
}

// ---------------- adj2[p][m][c] = Sum_i s1[i][m] * T_p[i][c] ----------------

__global__ void k_adj2(const float* __restrict__ s1, const float* __restrict__ T,
                       float* __restrict__ adj2) {
  int b = blockIdx.x;  // 768 = PP*16*16
  int p = b >> 8, rem = b & 255, m = rem >> 4, c = rem & 15;
  float acc = 0.f;
  for (int i = threadIdx.x; i < NN; i += 256)
    acc = fmaf(s1[(size_t)i * 16 + m], T[((size_t)p * NN + i) * 16 + c], acc);
  acc = blockReduceSum256(acc);
  if (threadIdx.x == 0) adj2[b] = acc;
}

// ---------------- Gram: G[m][c] = Sum_i s1[i][m]*s1[i][c] ----------------

__global__ void k_gram(const float* __restrict__ s1, float* __restrict__ G) {
  int b = blockIdx.x;  // 256
  int m = b >> 4, c = b & 15;
  float acc = 0.f;
  for (int i = threadIdx.x; i < NN; i += 256)
    acc = fmaf(s1[(size_t)i * 16 + m], s1[(size_t)i * 16 + c], acc);
  acc = blockReduceSum256(acc);
  if (threadIdx.x == 0) G[b] = acc;
}

// ---------------- x2[m][f] = Sum_n s1[n][m] * h[n][f] (h = elu(out2), F=12) ----------------

__global__ void k_x2(const float* __restrict__ s1, const float* __restrict__ h,
                     float* __restrict__ x2) {
  int b = blockIdx.x;  // 192 = 16*12
  int m = b / 12, f = b - m * 12;
  float acc = 0.f;
  for (int n = threadIdx.x; n < NN; n += 256)
    acc = fmaf(s1[(size_t)n * 16 + m], h[(size_t)n * 12 + f], acc);
  acc = blockReduceSum256(acc);
  if (threadIdx.x == 0) x2[b] = acc;
}

// ---------------- final: pool-2 (s2==1), penalties, MLP head ----------------

__global__ void k_final(const float* __restrict__ x2, const float* __restrict__ G,
                        const float* __restrict__ adj2, const float* __restrict__ scal,
                        const float* __restrict__ fc1w, const float* __restrict__ fc1b,
                        const float* __restrict__ fc2w, const float* __restrict__ fc2b,
                        const float* __restrict__ fc3w, const float* __restrict__ fc3b,
                        const float* __restrict__ fc4w, const float* __restrict__ fc4b,
                        float* __restrict__ out) {
  if (threadIdx.x != 0 || blockIdx.x != 0) return;
  // pool-2: softmax over a length-1 axis -> s2 = ones(16,1); x3 = column sums of x2
  float x3[12];
  for (int f = 0; f < 12; ++f) {
    float a = 0.f;
    for (int m = 0; m < 16; ++m) a += x2[m * 12 + f];
    x3[f] = a;
  }
  // r1 = sqrt(Sum adj^2 - 2*cross + P*||G||_F^2)/adj.size + mean entropy
  float gf = 0.f;
  for (int i = 0; i < 256; ++i) gf = fmaf(G[i], G[i], gf);
  float lsq = scal[0] - 2.f * scal[1] + (float)PP * gf;
  float r1 = sqrtf(fmaxf(lsq, 0.f)) / ((float)PP * (float)NN * (float)NN) + scal[3] / (float)NN;
  // r2: link = ||adj2 - ones||_F / (3*16*16); ent2 = -log(1+1e-15)
  float s2sq = 0.f;
  for (int i = 0; i < PP * 256; ++i) {
    float d = adj2[i] - 1.f;
    s2sq = fmaf(d, d, s2sq);
  }
  float r2 = sqrtf(s2sq) / (float)(PP * 256) + (-logf(1.f + 1e-15f));
  // MLP head
  float z1[20], z2[16], z3[20];
  for (int j = 0; j < 20; ++j) {
    float a = fc1b[j];
    for (int k = 0; k < 12; ++k) a = fmaf(x3[k], fc1w[k * 20 + j], a);
    z1[j] = eluf(a);
  }
  for (int j = 0; j < 16; ++j) {
    float a = fc2b[j];
    for (int k = 0; k < 20; ++k) a = fmaf(z1[k], fc2w[k * 16 + j], a);
    z2[j] = eluf(a);
  }
  for (int j = 0; j < 20; ++j) {
    float a = fc3b[j];
    for (int k = 0; k < 16; ++k) a = fmaf(z2[k], fc3w[k * 20 + j], a);
    z3[j] = eluf(a);
  }
  for (int j = 0; j < 2; ++j) {
    float a = fc4b[j];
    for (int k = 0; k < 20; ++k) a = fmaf(z3[k], fc4w[k * 2 + j], a);
    out[j] = a;
  }
  out[2] = r1 + r2;
}

// ---------------- launch ----------------

extern "C" void kernel_launch(void* const* d_in, const int* in_sizes, int n_in,
                              void* d_out, int out_size, void* d_ws, size_t ws_size,
                              hipStream_t stream) {
  const float* x   = (const float*)d_in[0];
  const int* eidx  = (const int*)d_in[1];
  const float* eat = (const float*)d_in[2];
  const float* adj = (const float*)d_in[3];
  const float* W1  = (const float*)d_in[4];
  const float* a1s = (const float*)d_in[5];
  const float* a1d = (const float*)d_in[6];
  const float* W2  = (const float*)d_in[7];
  const float* a2s = (const float*)d_in[8];
  const float* a2d = (const float*)d_in[9];
  const float* S1  = (const float*)d_in[10];
  // d_in[11] = S2 : softmax over axis of size 1 -> all ones, unused
  const float* fc1w = (const float*)d_in[12];
  const float* fc1b = (const float*)d_in[13];
  const float* fc2w = (const float*)d_in[14];
  const float* fc2b = (const float*)d_in[15];
  const float* fc3w = (const float*)d_in[16];
  const float* fc3b = (const float*)d_in[17];
  const float* fc4w = (const float*)d_in[18];
  const float* fc4b = (const float*)d_in[19];
  float* out = (float*)d_out;

  const int* src = eidx;       // edge_index[0]
  const int* dst = eidx + EE;  // edge_index[1]

  float* W = (float*)d_ws;
  size_t o = 0;
  float* h1    = W + o; o += (size_t)NN * 16;
  float* ssrc1 = W + o; o += NN;
  float* sdst1 = W + o; o += NN;
  float* smax1 = W + o; o += NN;
  float* den1  = W + o; o += NN;
  float* ex1   = W + o; o += EE;
  float* ae1   = W + o; o += (size_t)EE * PP;
  float* out1  = W + o; o += (size_t)NN * 48;
  float* h2    = W + o; o += (size_t)NN * 4;
  float* ssrc2 = W + o; o += NN;
  float* sdst2 = W + o; o += NN;
  float* smax2 = W + o; o += NN;
  float* den2  = W + o; o += NN;
  float* ex2   = W + o; o += EE;
  float* ae2   = W + o; o += (size_t)EE * PP;
  float* out2  = W + o; o += (size_t)NN * 12;
  float* s1    = W + o; o += (size_t)NN * 16;
  float* s1T   = W + o; o += (size_t)NN * 16;
  float* T     = W + o; o += (size_t)PP * NN * 16;
  float* Tpart = W + o; o += (size_t)KSPLIT * PP * NN * 16;
  float* x2    = W + o; o += 192;
  float* adj2  = W + o; o += 768;
  float* G     = W + o; o += 256;
  float* scal  = W + o; o += 16;  // [0]=Sum adj^2, [1]=cross, [3]=entropy sum
  int* deg     = (int*)(W + o); o += NN;
  int* offs    = (int*)(W + o); o += NN + 1;
  int* cursor  = (int*)(W + o); o += NN;
  int* eperm   = (int*)(W + o); o += EE;

  const int TB = 256;
  auto g = [](int n, int b) { return (n + b - 1) / b; };

  // --- init accumulators (re-run every call: deterministic work) ---
  k_fill<<<g(NN, TB), TB, 0, stream>>>(smax1, -3.4e38f, NN);
  k_fill<<<g(NN, TB), TB, 0, stream>>>(smax2, -3.4e38f, NN);
  k_fill<<<g(NN, TB), TB, 0, stream>>>(den1, 0.f, NN);
  k_fill<<<g(NN, TB), TB, 0, stream>>>(den2, 0.f, NN);
  k_fill<<<1, 16, 0, stream>>>(scal, 0.f, 16);
  k_filli<<<g(NN, TB), TB, 0, stream>>>(deg, 0, NN);

  // --- CSR by destination (shared by both EGAT layers) ---
  k_count<<<g(EE, TB), TB, 0, stream>>>(dst, deg);
  k_scan4096<<<1, 256, 0, stream>>>(deg, offs, cursor);
  k_scatter<<<g(EE, TB), TB, 0, stream>>>(dst, cursor, eperm);

  // --- EGAT layer 1 ---
  k_gemm1_wmma<<<NN / 16, 32, 0, stream>>>(x, W1, h1);
  k_nodeproj<<<g(NN, TB), TB, 0, stream>>>(h1, 16, a1s, a1d, ssrc1, sdst1);
  k_score_max<<<g(EE, TB), TB, 0, stream>>>(src, dst, ssrc1, sdst1, ex1, smax1);
  k_expden<<<g(EE, TB), TB, 0, stream>>>(dst, ex1, smax1, den1);
  k_ae<<<g(EE * PP, TB), TB, 0, stream>>>(dst, ex1, den1, eat, ae1);
  k_gather_msg<<<NN, 64, 0, stream>>>(offs, eperm, src, ae1, h1, out1, 16);
  k_elu<<<g(NN * 48, TB), TB, 0, stream>>>(out1, NN * 48);

  // --- EGAT layer 2 ---
  k_gemm2<<<g(NN * 4, TB), TB, 0, stream>>>(out1, W2, h2);
  k_nodeproj<<<g(NN, TB), TB, 0, stream>>>(h2, 4, a2s, a2d, ssrc2, sdst2);
  k_score_max<<<g(EE, TB), TB, 0, stream>>>(src, dst, ssrc2, sdst2, ex2, smax2);
  k_expden<<<g(EE, TB), TB, 0, stream>>>(dst, ex2, smax2, den2);
  k_ae<<<g(EE * PP, TB), TB, 0, stream>>>(dst, ex2, den2, ae1, ae2);
  k_gather_msg<<<NN, 32, 0, stream>>>(offs, eperm, src, ae2, h2, out2, 4);
  k_elu<<<g(NN * 12, TB), TB, 0, stream>>>(out2, NN * 12);

  // --- DiffPool 1 (single pass over 201 MB adj, WMMA, K split 8-way) ---
  k_softmax_s1<<<g(NN, TB), TB, 0, stream>>>(S1, s1, &scal[3]);
  k_transpose_s1<<<g(NN * 16, TB), TB, 0, stream>>>(s1, s1T);
  k_adj_pool<<<dim3(NN / 16, PP, KSPLIT), 32, 0, stream>>>(adj, s1T, Tpart, &scal[0]);
  k_reduceT<<<g(PP * NN * 16, TB), TB, 0, stream>>>(Tpart, T);
  k_cross<<<g(PP * NN * 16, TB), TB, 0, stream>>>(s1, T, &scal[1]);
  k_adj2<<<PP * 256, TB, 0, stream>>>(s1, T, adj2);
  k_gram<<<256, TB, 0, stream>>>(s1, G);
  k_x2<<<192, TB, 0, stream>>>(s1, out2, x2);

  // --- DiffPool 2 (degenerate) + penalties + MLP head ---
  k_final<<<1, 1, 0, stream>>>(x2, G, adj2, scal, fc1w, fc1b, fc2w, fc2b, fc3w, fc3b,
                               fc4w, fc4b, out);
}